// frozenInputToLeaf_52149492908459
// MI455X (gfx1250) — compile-verified
//
#include <hip/hip_runtime.h>
#include <stdint.h>
#include <stddef.h>

#define BATCH          16384
#define NUM_INPUTS     4096
#define NUM_LEAVES     8192
#define ROWS_PER_BLOCK 8
#define THREADS        1024
#define LDS_BYTES      (ROWS_PER_BLOCK * NUM_INPUTS * 4)   // 128 KB

typedef unsigned int u32x4 __attribute__((ext_vector_type(4)));
typedef int          i32x8 __attribute__((ext_vector_type(8)));
typedef int          i32x4 __attribute__((ext_vector_type(4)));
typedef float        f32x4 __attribute__((ext_vector_type(4)));

// amdgpu-toolchain (therock-10.0 headers) ships the TDM header and uses the
// 6-arg builtin; ROCm 7.2 clang-22 uses the 5-arg form (probe-confirmed).
#if defined(__HIP_DEVICE_COMPILE__) && __has_include(<hip/amd_detail/amd_gfx1250_TDM.h>)
#define TDM_SIX_ARGS 1
#else
#define TDM_SIX_ARGS 0
#endif

__device__ __forceinline__ void wait_tensorcnt0() {
#if __has_builtin(__builtin_amdgcn_s_wait_tensorcnt)
  __builtin_amdgcn_s_wait_tensorcnt(0);
#else
  asm volatile("s_wait_tensorcnt 0" ::: "memory");
#endif
}

__global__ void __launch_bounds__(THREADS)
gather_tdm_kernel(const float* __restrict__ x,
                  const long long* __restrict__ idx,
                  float* __restrict__ out) {
  extern __shared__ float xs[];          // ROWS_PER_BLOCK x 4096 f32 at LDS offset 0

  const unsigned t    = threadIdx.x;
  const unsigned row0 = blockIdx.x * ROWS_PER_BLOCK;

  // ---------- Stage x tile (8 rows x 16 KB) into LDS ----------
#if __has_builtin(__builtin_amdgcn_tensor_load_to_lds)
  // Single TDM DMA for the whole tile, issued by wave 0 only.
  // TDM ignores EXEC, so the guard must be a *scalar* branch: readfirstlane
  // makes the condition provably wave-uniform -> s_cmp + s_cbranch.
  const unsigned wid = ((unsigned)__builtin_amdgcn_readfirstlane((int)t)) >> 5;
  if (wid == 0) {
    const unsigned long long gaddr =
        (unsigned long long)(uintptr_t)x +
        (unsigned long long)row0 * (NUM_INPUTS * 4ull);

    u32x4 g0;
    g0[0] = 1u;                                             // count=1 (valid D#)
    g0[1] = 0u;                                             // lds_addr = 0 (xs base)
    g0[2] = (unsigned)(gaddr & 0xFFFFFFFFull);              // global_addr[31:0]
    g0[3] = (unsigned)((gaddr >> 32) & 0x1FFFFFFull)        // global_addr[56:32]
          | (2u << 30);                                     // type = 2 ("image")

    i32x8 g1;
    g1[0] = (int)(2u << 16);                                // data_size = 4 bytes
    g1[1] = (int)((unsigned)(NUM_INPUTS & 0xFFFF) << 16);   // tensor_dim0[15:0]
    g1[2] = (int)((unsigned)ROWS_PER_BLOCK << 16);          // tensor_dim1[15:0]
    g1[3] = (int)((unsigned)(NUM_INPUTS & 0xFFFF) << 16);   // tile_dim0
    g1[4] = (int)ROWS_PER_BLOCK;                            // tile_dim1 (tile_dim2=0)
    g1[5] = (int)NUM_INPUTS;                                // tensor_dim0_stride[31:0]
    g1[6] = 0;
    g1[7] = 0;

    i32x4 gz = {0, 0, 0, 0};                                // 2D tensor: groups 2/3 unused
#if TDM_SIX_ARGS
    i32x8 gz8 = {0, 0, 0, 0, 0, 0, 0, 0};
    __builtin_amdgcn_tensor_load_to_lds(g0, g1, gz, gz, gz8, 0);
#else
    __builtin_amdgcn_tensor_load_to_lds(g0, g1, gz, gz, 0);
#endif
  }
#else
  {
    // Fallback: cooperative b128 copy (also what the host pass parses).
    const f32x4* src4 = (const f32x4*)(x + (size_t)row0 * NUM_INPUTS);
    f32x4*       dst4 = (f32x4*)xs;
    for (unsigned i = t; i < (ROWS_PER_BLOCK * NUM_INPUTS) / 4; i += THREADS)
      dst4[i] = src4[i];
  }
#endif

  // ---------- Overlap with the DMA: fetch this thread's 8 column indices ----------
  const unsigned c0 = 4u * t;            // columns [c0, c0+3]
  const unsigned c1 = c0 + 4u * THREADS; // columns [c0+4096, ...]
  int ia[4], ib[4];
#pragma unroll
  for (int j = 0; j < 4; ++j) {
    ia[j] = (int)idx[c0 + j];            // int64 indices, values < 4096
    ib[j] = (int)idx[c1 + j];
  }

#if __has_builtin(__builtin_amdgcn_tensor_load_to_lds)
  wait_tensorcnt0();                     // no-op for waves with TENSORcnt==0
#endif
  __syncthreads();                       // LDS tile visible to all waves

  // ---------- Gather from LDS banks, stream NT b128 stores ----------
#pragma unroll
  for (int r = 0; r < ROWS_PER_BLOCK; ++r) {
    const float* rowp = xs + (r << 12);  // r * 4096
    f32x4 va, vb;
#pragma unroll
    for (int j = 0; j < 4; ++j) {
      va[j] = rowp[ia[j]];               // ds_load_b32 gathers
      vb[j] = rowp[ib[j]];
    }
    float* orow = out + (size_t)(row0 + r) * NUM_LEAVES;
    __builtin_nontemporal_store(va, (f32x4*)(orow + c0));  // write-once stream
    __builtin_nontemporal_store(vb, (f32x4*)(orow + c1));
  }
}

extern "C" void kernel_launch(void* const* d_in, const int* in_sizes, int n_in,
                              void* d_out, int out_size, void* d_ws, size_t ws_size,
                              hipStream_t stream) {
  const float*     x   = (const float*)d_in[0];
  const long long* idx = (const long long*)d_in[1];
  float*           out = (float*)d_out;

  (void)in_sizes; (void)n_in; (void)out_size; (void)d_ws; (void)ws_size;

  // 128 KB dynamic LDS per workgroup (320 KB available per WGP on CDNA5).
  (void)hipFuncSetAttribute((const void*)gather_tdm_kernel,
                            hipFuncAttributeMaxDynamicSharedMemorySize,
                            LDS_BYTES);

  dim3 grid(BATCH / ROWS_PER_BLOCK);     // 2048 blocks
  gather_tdm_kernel<<<grid, THREADS, LDS_BYTES, stream>>>(x, idx, out);
}